// BlockDiagonal_76974403879325
// MI455X (gfx1250) — compile-verified
//
#include <hip/hip_runtime.h>

// Block-diagonal fused GEMM for MI455X (gfx1250):
//   Y[n, b*256 + o] = sum_k X[n, b*256 + k] * W[b, o, k] + bias[b, o]
// BATCH=16384, NUM_BLOCKS=16, BLOCK_IN=BLOCK_OUT=256, all fp32.
//
// Roofline: 34.4 GFLOP vs ~540 MB mandatory HBM traffic -> memory-bound
// (~23 us floor @ 23.3 TB/s). fp32 WMMA (V_WMMA_F32_16X16X4_F32) keeps full
// precision; compute is nowhere near the limiter. Y stores are non-temporal
// so the write-once output doesn't evict the L2-resident W (4 MB) or X tiles.

typedef float v2f __attribute__((ext_vector_type(2)));
typedef float v8f __attribute__((ext_vector_type(8)));

#define BATCH      16384
#define NB         16
#define BIN        256
#define BOUT       256
#define IN_FEAT    4096
#define OUT_FEAT   4096

__global__ __launch_bounds__(256) void
bd_gemm_wmma_f32(const float* __restrict__ X,
                 const float* __restrict__ W,
                 const float* __restrict__ Bias,
                 float* __restrict__ Y)
{
    const int lane = threadIdx.x & 31;
    const int wave = threadIdx.x >> 5;   // 0..7
    const int wm   = wave & 1;           // 2 M-subtiles
    const int wn   = wave >> 1;          // 4 N-subtiles of 64 cols
    const int blk  = blockIdx.y;         // which diagonal block (0..15)
    const int m0   = blockIdx.x * 32 + wm * 16;
    const int n0   = wn * 64;

    const int mL = lane & 15;            // M index (A) / N index (B,C,D)
    const int hL = lane >> 4;            // lane half selects K pair / M+8

    // A: row of X for this lane, within this block's K-slice
    const float* xRow = X + (size_t)(m0 + mL) * IN_FEAT + (size_t)blk * BIN;
    // B: W[blk, n, k] row-major in k -> per-lane float2 along K is contiguous
    const float* wBlk = W + (size_t)blk * BOUT * BIN;
    const float* wR0  = wBlk + (size_t)(n0 +  0 + mL) * BIN;
    const float* wR1  = wBlk + (size_t)(n0 + 16 + mL) * BIN;
    const float* wR2  = wBlk + (size_t)(n0 + 32 + mL) * BIN;
    const float* wR3  = wBlk + (size_t)(n0 + 48 + mL) * BIN;

    v8f acc0 = {}, acc1 = {}, acc2 = {}, acc3 = {};

    const int kLane = 2 * hL;            // this lane's K offset within a 4-step

#pragma unroll 8
    for (int k0 = 0; k0 < BIN; k0 += 4) {
        v2f a  = *(const v2f*)(xRow + k0 + kLane);
        v2f b0 = *(const v2f*)(wR0  + k0 + kLane);
        v2f b1 = *(const v2f*)(wR1  + k0 + kLane);
        v2f b2 = *(const v2f*)(wR2  + k0 + kLane);
        v2f b3 = *(const v2f*)(wR3  + k0 + kLane);

        // 8 args: (neg_a, A, neg_b, B, c_mod, C, reuse_a, reuse_b)
        acc0 = __builtin_amdgcn_wmma_f32_16x16x4_f32(false, a, false, b0,
                                                     (short)0, acc0, false, false);
        acc1 = __builtin_amdgcn_wmma_f32_16x16x4_f32(false, a, false, b1,
                                                     (short)0, acc1, false, false);
        acc2 = __builtin_amdgcn_wmma_f32_16x16x4_f32(false, a, false, b2,
                                                     (short)0, acc2, false, false);
        acc3 = __builtin_amdgcn_wmma_f32_16x16x4_f32(false, a, false, b3,
                                                     (short)0, acc3, false, false);
    }

    // Bias: depends only on the lane's N index
    const float bias0 = Bias[(size_t)blk * BOUT + n0 +  0 + mL];
    const float bias1 = Bias[(size_t)blk * BOUT + n0 + 16 + mL];
    const float bias2 = Bias[(size_t)blk * BOUT + n0 + 32 + mL];
    const float bias3 = Bias[(size_t)blk * BOUT + n0 + 48 + mL];

    // C/D layout: VGPR v, lanes 0-15 -> M=v, lanes 16-31 -> M=v+8; N = lane&15
    // Y is write-once: non-temporal stores keep it from thrashing L2.
    const size_t colBase = (size_t)blk * BOUT + n0 + mL;
#pragma unroll
    for (int v = 0; v < 8; ++v) {
        const size_t row = (size_t)(m0 + v + 8 * hL);
        float* yRow = Y + row * OUT_FEAT + colBase;
        __builtin_nontemporal_store(acc0[v] + bias0, yRow +  0);
        __builtin_nontemporal_store(acc1[v] + bias1, yRow + 16);
        __builtin_nontemporal_store(acc2[v] + bias2, yRow + 32);
        __builtin_nontemporal_store(acc3[v] + bias3, yRow + 48);
    }
}

extern "C" void kernel_launch(void* const* d_in, const int* in_sizes, int n_in,
                              void* d_out, int out_size, void* d_ws, size_t ws_size,
                              hipStream_t stream)
{
    const float* X    = (const float*)d_in[0];
    const float* W    = (const float*)d_in[1];
    const float* Bias = (const float*)d_in[2];
    float*       Y    = (float*)d_out;

    dim3 grid(BATCH / 32, NB);   // 512 row-tiles x 16 blocks = 8192 WGs
    dim3 block(256);             // 8 wave32s: 2(M) x 4(N)
    bd_gemm_wmma_f32<<<grid, block, 0, stream>>>(X, W, Bias, Y);
}